// LGN_single_10746008175455
// MI455X (gfx1250) — compile-verified
//
#include <hip/hip_runtime.h>
#include <hip/hip_bf16.h>

// ---------------------------------------------------------------------------
// LightGCN propagation for MI455X (gfx1250, wave32).
//  * counting-sort edges to CSR once (no fp32 atomics in the hot loop)
//  * SpMM: one wave32 per row; 32 lanes x float2 => every gather is one
//    256B-coalesced wave load; embedding table (38MB) is L2-resident.
//  * gfx1250 paths: GLOBAL_LOAD_ASYNC_TO_LDS_B64 stages each row's packed
//    (col,val) edge segment into LDS (ASYNCcnt decoupled from LOADcnt
//    gathers), s_wait_asynccnt sync, global_prefetch_b8 gather prefetch.
//  * fused: layer accumulation into d_out, final /4, passthrough copies.
// ---------------------------------------------------------------------------

namespace {
constexpr int kUsers   = 100000;
constexpr int kItems   = 50000;
constexpr int kN       = kUsers + kItems;   // 150000
constexpr int kD       = 64;
constexpr int kE       = 2400000;
constexpr int kScanB   = 1024;
constexpr int kEdgeCap = 64;                // per-row LDS staging capacity
}

// ---------------------------------------------------------------------------

__global__ void zero_i32(int* __restrict__ p, int n) {
  int i = blockIdx.x * blockDim.x + threadIdx.x;
  if (i < n) p[i] = 0;
}

// embA = concat(user, item); light(d_out) = same; also emit user/item output
// copies in the same pass. All float4-vectorized.
__global__ void init_embed(const float4* __restrict__ user4,
                           const float4* __restrict__ item4,
                           float4* __restrict__ embA4,
                           float4* __restrict__ light4,
                           float4* __restrict__ uo4,
                           float4* __restrict__ io4) {
  int i = blockIdx.x * blockDim.x + threadIdx.x;
  constexpr int total4 = kN * kD / 4;
  constexpr int u4     = kUsers * kD / 4;
  if (i >= total4) return;
  float4 v = (i < u4) ? user4[i] : item4[i - u4];
  embA4[i]  = v;
  light4[i] = v;
  if (i < u4) uo4[i] = v;
  else        io4[i - u4] = v;
}

__global__ void edge_hist(const int* __restrict__ rows, int* __restrict__ cnt) {
  int e = blockIdx.x * blockDim.x + threadIdx.x;
  if (e < kE) atomicAdd(&cnt[rows[e]], 1);
}

// 3-phase exclusive scan over kN counts.
__global__ void scan_pass1(const int* __restrict__ cnt, int* __restrict__ excl,
                           int* __restrict__ aux, int n) {
  __shared__ int s[kScanB];
  int tid = threadIdx.x;
  int g   = blockIdx.x * kScanB + tid;
  int v   = (g < n) ? cnt[g] : 0;
  s[tid]  = v;
  __syncthreads();
  for (int off = 1; off < kScanB; off <<= 1) {
    int t = (tid >= off) ? s[tid - off] : 0;
    __syncthreads();
    s[tid] += t;
    __syncthreads();
  }
  if (g < n) excl[g] = s[tid] - v;           // exclusive partial
  if (tid == kScanB - 1) aux[blockIdx.x] = s[tid];
}

__global__ void scan_pass2(int* __restrict__ aux, int n) {
  __shared__ int s[kScanB];
  int tid = threadIdx.x;
  int v   = (tid < n) ? aux[tid] : 0;
  s[tid]  = v;
  __syncthreads();
  for (int off = 1; off < kScanB; off <<= 1) {
    int t = (tid >= off) ? s[tid - off] : 0;
    __syncthreads();
    s[tid] += t;
    __syncthreads();
  }
  if (tid < n) aux[tid] = s[tid] - v;        // exclusive in place
}

__global__ void scan_pass3(int* __restrict__ rowp, int* __restrict__ cur,
                           const int* __restrict__ aux, int n) {
  int g = blockIdx.x * kScanB + threadIdx.x;
  if (g < n) {
    int val = rowp[g] + aux[blockIdx.x];
    rowp[g] = val;
    cur[g]  = val;                           // scatter cursor = row starts
  }
  if (g == 0) rowp[n] = kE;
}

// Pack (col, val) into one 8B element so the hot loop touches a single
// stream.
__global__ void edge_scatter(const int* __restrict__ rows,
                             const int* __restrict__ cols,
                             const float* __restrict__ vals,
                             int* __restrict__ cur,
                             int2* __restrict__ sedge) {
  int e = blockIdx.x * blockDim.x + threadIdx.x;
  if (e >= kE) return;
  int r = rows[e];
  int p = atomicAdd(&cur[r], 1);
  sedge[p] = make_int2(cols[e], __float_as_int(vals[e]));
}

// One wave32 per row. lane L owns dims [2L, 2L+1]. The row's packed edge
// segment is staged into a per-wave LDS slot with GLOBAL_LOAD_ASYNC_TO_LDS_B64
// (one instruction = 32 edges = 256B), synced with s_wait_asynccnt, then the
// FMA loop broadcasts edges from LDS while 256B embedding gathers stream on
// LOADcnt with global_prefetch_b8 lookahead. Long rows (> kEdgeCap) fall back
// to uniform scalar-path loads. Fuses light += emb and *0.25 on last layer.
__global__ void __launch_bounds__(256) spmm_rows(
    const int* __restrict__ rowp, const int2* __restrict__ sedge,
    const float* __restrict__ xin, float* __restrict__ xout,
    float* __restrict__ light, int last) {
  __shared__ int2 ebuf[8][kEdgeCap];         // 8 waves * 64 edges * 8B = 4KB
  int wv  = threadIdx.x >> 5;
  int row = __builtin_amdgcn_readfirstlane((int)(blockIdx.x * 8u + wv));
  if (row >= kN) return;
  int lane = threadIdx.x & 31;
  int d0   = lane * 2;

  int beg = __builtin_amdgcn_readfirstlane(rowp[row]);
  int end = __builtin_amdgcn_readfirstlane(rowp[row + 1]);
  int nE  = end - beg;

  float ax = 0.0f, ay = 0.0f;

  if (nE <= kEdgeCap) {
    // --- async stage this row's edges into the wave's LDS slot -------------
    if (lane < nE) {
      unsigned    ldsOff = (unsigned)(size_t)(void*)&ebuf[wv][lane];
      const int2* ga     = sedge + beg + lane;
      asm volatile("global_load_async_to_lds_b64 %0, %1, off"
                   :
                   : "v"(ldsOff), "v"(ga)
                   : "memory");
    }
    {
      int i2 = lane + 32;
      if (i2 < nE) {
        unsigned    ldsOff = (unsigned)(size_t)(void*)&ebuf[wv][i2];
        const int2* ga     = sedge + beg + i2;
        asm volatile("global_load_async_to_lds_b64 %0, %1, off"
                     :
                     : "v"(ldsOff), "v"(ga)
                     : "memory");
      }
    }
    asm volatile("s_wait_asynccnt 0" ::: "memory");

    // --- FMA loop: LDS edge broadcast + coalesced 256B gathers -------------
    for (int i = 0; i < nE; ++i) {
      int2  ev = ebuf[wv][i];
      int   c  = __builtin_amdgcn_readfirstlane(ev.x);
      float v  = __int_as_float(__builtin_amdgcn_readfirstlane(ev.y));
      if (i + 4 < nE) {
        int cp = __builtin_amdgcn_readfirstlane(ebuf[wv][i + 4].x);
        __builtin_prefetch(xin + (size_t)cp * kD + d0, 0, 0);
      }
      const float2 x = *(const float2*)(xin + (size_t)c * kD + d0);
      ax = fmaf(v, x.x, ax);
      ay = fmaf(v, x.y, ay);
    }
  } else {
    // --- fallback: uniform-address (scalar path) edge loads ----------------
    for (int e = beg; e < end; ++e) {
      int2  ev = sedge[e];
      int   c  = __builtin_amdgcn_readfirstlane(ev.x);
      float v  = __int_as_float(__builtin_amdgcn_readfirstlane(ev.y));
      const float2 x = *(const float2*)(xin + (size_t)c * kD + d0);
      ax = fmaf(v, x.x, ax);
      ay = fmaf(v, x.y, ay);
    }
  }

  size_t o = (size_t)row * kD + d0;
  *(float2*)(xout + o) = make_float2(ax, ay);

  float2 lv = *(const float2*)(light + o);
  lv.x += ax;
  lv.y += ay;
  if (last) { lv.x *= 0.25f; lv.y *= 0.25f; }
  *(float2*)(light + o) = lv;
}

// ---------------------------------------------------------------------------

extern "C" void kernel_launch(void* const* d_in, const int* in_sizes, int n_in,
                              void* d_out, int out_size, void* d_ws,
                              size_t ws_size, hipStream_t stream) {
  const float* user_emb = (const float*)d_in[0];
  const float* item_emb = (const float*)d_in[1];
  const float* gvals    = (const float*)d_in[2];
  const int*   grows    = (const int*)d_in[3];
  const int*   gcols    = (const int*)d_in[4];

  // Output layout: light_out [kN*kD] | user_emb [kUsers*kD] | item_emb [kItems*kD]
  float* light = (float*)d_out;
  float* uo    = light + (size_t)kN * kD;
  float* io    = uo + (size_t)kUsers * kD;

  // Workspace carve (~97 MB).
  char* w = (char*)d_ws;
  auto take = [&](size_t bytes) -> char* {
    char* p = w;
    w += (bytes + 255) & ~(size_t)255;
    return p;
  };
  float* embA  = (float*)take((size_t)kN * kD * 4);
  float* embB  = (float*)take((size_t)kN * kD * 4);
  int2*  sedge = (int2*)take((size_t)kE * 8);
  int*   cnt   = (int*)take((size_t)(kN + 1) * 4);
  int*   rowp  = (int*)take((size_t)(kN + 2) * 4);
  int*   cur   = (int*)take((size_t)(kN + 1) * 4);
  int*   aux   = (int*)take((size_t)kScanB * 4);
  (void)ws_size; (void)in_sizes; (void)n_in; (void)out_size;

  const int nScan = (kN + kScanB - 1) / kScanB;  // 147 (<= 1024)

  // 1) zero histogram
  zero_i32<<<(kN + 1 + 255) / 256, 256, 0, stream>>>(cnt, kN + 1);

  // 2) build all_emb, seed accumulator, emit passthrough outputs
  {
    constexpr int total4 = kN * kD / 4;
    init_embed<<<(total4 + 255) / 256, 256, 0, stream>>>(
        (const float4*)user_emb, (const float4*)item_emb, (float4*)embA,
        (float4*)light, (float4*)uo, (float4*)io);
  }

  // 3) counting sort -> CSR (packed edges)
  edge_hist<<<(kE + 255) / 256, 256, 0, stream>>>(grows, cnt);
  scan_pass1<<<nScan, kScanB, 0, stream>>>(cnt, rowp, aux, kN);
  scan_pass2<<<1, kScanB, 0, stream>>>(aux, nScan);
  scan_pass3<<<nScan, kScanB, 0, stream>>>(rowp, cur, aux, kN);
  edge_scatter<<<(kE + 255) / 256, 256, 0, stream>>>(grows, gcols, gvals, cur,
                                                     sedge);

  // 4) three fused SpMM + accumulate layers (last fuses the /4)
  const float* xin = embA;
  float*       xo  = embB;
  const int spmmBlocks = (kN + 7) / 8;  // 8 waves (rows) per 256-thread block
  for (int layer = 0; layer < 3; ++layer) {
    spmm_rows<<<spmmBlocks, 256, 0, stream>>>(rowp, sedge, xin, xo, light,
                                              layer == 2 ? 1 : 0);
    const float* t = xin;
    xin = xo;
    xo  = (float*)t;
  }
}